// PoemGenerater_64458869178956
// MI455X (gfx1250) — compile-verified
//
#include <hip/hip_runtime.h>
#include <hip/hip_bf16.h>
#include <math.h>

// ---------------- model constants ----------------
#define EMB   256
#define POSD  64
#define HH    512        // encoder hidden per direction
#define VV    12000
#define NLAY  2
#define NSENT 4
#define NCHR  7
#define LSEQ  32
#define BB    128
#define H2    1024       // 2H (decoder hidden)
#define H4    2048       // 4H

typedef float v2f __attribute__((ext_vector_type(2)));
typedef float v8f __attribute__((ext_vector_type(8)));

// =====================================================================
// fp32 WMMA GEMM:  C[M,N] = A[M,K]*W[N,K]^T (+ A2[M,K2]*W2[N,K2]^T) (+bias) (+acc C)
// Each wave computes a 16x64 strip: 4 accumulators share one A fragment.
// K must be a multiple of 8 (true for all call sites: 256/512/1024/1344/2048).
// The K-loop is split into an explicit 10-load clause followed by 8 WMMAs on
// 4 independent accumulator chains so loads overlap the matrix pipe.
// Fragment layouts per CDNA5 ISA 7.12.2:
//   A 16x4 : lanes 0-15 hold M=0..15 K={0,1}; lanes 16-31 K={2,3}
//   B 4x16 : VGPR v -> K=v (lanes 0-15), K=v+2 (lanes 16-31); N = lane%16
//   C/D    : VGPR r -> M=r (lanes 0-15), M=r+8 (lanes 16-31); N = lane%16
// =====================================================================
__device__ __forceinline__ void wmma_k8(const float* __restrict__ Arow,
                                        const float* const Wrow[4],
                                        int kk, v8f c[4])
{
    v2f a0 = *(const v2f*)(Arow + kk);
    v2f a1 = *(const v2f*)(Arow + kk + 4);
    v2f b00 = *(const v2f*)(Wrow[0] + kk);
    v2f b01 = *(const v2f*)(Wrow[1] + kk);
    v2f b02 = *(const v2f*)(Wrow[2] + kk);
    v2f b03 = *(const v2f*)(Wrow[3] + kk);
    v2f b10 = *(const v2f*)(Wrow[0] + kk + 4);
    v2f b11 = *(const v2f*)(Wrow[1] + kk + 4);
    v2f b12 = *(const v2f*)(Wrow[2] + kk + 4);
    v2f b13 = *(const v2f*)(Wrow[3] + kk + 4);
    // (neg_a, A, neg_b, B, c_mod, C, reuse_a, reuse_b)
    c[0] = __builtin_amdgcn_wmma_f32_16x16x4_f32(false, a0, false, b00, (short)0, c[0], false, false);
    c[1] = __builtin_amdgcn_wmma_f32_16x16x4_f32(false, a0, false, b01, (short)0, c[1], false, false);
    c[2] = __builtin_amdgcn_wmma_f32_16x16x4_f32(false, a0, false, b02, (short)0, c[2], false, false);
    c[3] = __builtin_amdgcn_wmma_f32_16x16x4_f32(false, a0, false, b03, (short)0, c[3], false, false);
    c[0] = __builtin_amdgcn_wmma_f32_16x16x4_f32(false, a1, false, b10, (short)0, c[0], false, false);
    c[1] = __builtin_amdgcn_wmma_f32_16x16x4_f32(false, a1, false, b11, (short)0, c[1], false, false);
    c[2] = __builtin_amdgcn_wmma_f32_16x16x4_f32(false, a1, false, b12, (short)0, c[2], false, false);
    c[3] = __builtin_amdgcn_wmma_f32_16x16x4_f32(false, a1, false, b13, (short)0, c[3], false, false);
}

__global__ __launch_bounds__(128)
void k_gemm(const float* __restrict__ A, int lda, int K,
            const float* __restrict__ W, int ldw,
            const float* __restrict__ A2, int lda2, int K2,
            const float* __restrict__ W2, int ldw2,
            const float* __restrict__ bias,
            float* __restrict__ C, int ldc,
            int M, int N, int acc)
{
    const int lane = threadIdx.x & 31;
    const int wave = threadIdx.x >> 5;
    const int n0 = (blockIdx.x * 4 + wave) * 64;
    const int m0 = blockIdx.y * 16;
    if (n0 >= N || m0 >= M) return;          // wave-uniform: EXEC stays all-ones

    const int mr  = lane & 15;               // row in A-tile / col in W
    const int khi = lane >> 4;               // 0 or 1
    const int klo = khi * 2;                 // K offset within step-of-4

    const float* Arow = A + (size_t)(m0 + mr) * lda + klo;

    int cb[4];
    const float* Wrow[4];
#pragma unroll
    for (int t = 0; t < 4; ++t) {
        const int c0 = n0 + t * 16;
        cb[t] = (c0 + 16 <= N) ? c0 : (N - 16);   // clamp: safe loads for partial strips
        Wrow[t] = W + (size_t)(cb[t] + mr) * ldw + klo;
    }

    v8f c[4];
#pragma unroll
    for (int t = 0; t < 4; ++t) {
        const int col = cb[t] + mr;
        const float bv = bias ? bias[col] : 0.0f;
#pragma unroll
        for (int r = 0; r < 8; ++r) {
            const int row = m0 + r + (khi << 3);
            float cv = bv;
            if (acc) cv += C[(size_t)row * ldc + col];
            c[t][r] = cv;
        }
    }

    for (int kk = 0; kk < K; kk += 8)
        wmma_k8(Arow, Wrow, kk, c);

    if (A2) {  // fused second contribution (e.g. h @ Whh^T)
        const float* Arow2 = A2 + (size_t)(m0 + mr) * lda2 + klo;
        const float* Wrow2[4];
#pragma unroll
        for (int t = 0; t < 4; ++t)
            Wrow2[t] = W2 + (size_t)(cb[t] + mr) * ldw2 + klo;
        for (int kk = 0; kk < K2; kk += 8)
            wmma_k8(Arow2, Wrow2, kk, c);
    }

#pragma unroll
    for (int t = 0; t < 4; ++t) {
        if (n0 + t * 16 >= N) continue;      // wave-uniform partial-strip guard
        const int col = cb[t] + mr;
#pragma unroll
        for (int r = 0; r < 8; ++r) {
            const int row = m0 + r + (khi << 3);
            C[(size_t)row * ldc + col] = c[t][r];
        }
    }
}

// ---------------- pointwise LSTM cell ----------------
// z: (B, 4*Hh) gate order i,f,g,o ; cprev/h/cout: (B,Hh) contiguous.
// Optionally scatter h into h2[b*h2ld + j] (encoder fwd/bwd halves).
__global__ __launch_bounds__(256)
void k_cell(const float* __restrict__ z, int zld,
            const float* __restrict__ cprev,
            float* __restrict__ h, float* __restrict__ cout,
            int Hh, float* __restrict__ h2, int h2ld)
{
    int i = blockIdx.x * blockDim.x + threadIdx.x;
    if (i >= BB * Hh) return;
    const int b = i / Hh, j = i - b * Hh;
    const float* zr = z + (size_t)b * zld;
    const float zi = zr[j];
    const float zf = zr[Hh + j];
    const float zg = zr[2 * Hh + j];
    const float zo = zr[3 * Hh + j];
    const float si = 1.0f / (1.0f + expf(-zi));
    const float sf = 1.0f / (1.0f + expf(-zf));
    const float so = 1.0f / (1.0f + expf(-zo));
    const float cn = sf * cprev[i] + si * tanhf(zg);
    const float hn = so * tanhf(cn);
    cout[i] = cn;
    h[i]    = hn;
    if (h2) h2[(size_t)b * h2ld + j] = hn;
}

// ---------------- misc elementwise kernels ----------------
__global__ void k_fill0(float* p, int n)
{
    int i = blockIdx.x * blockDim.x + threadIdx.x;
    if (i < n) p[i] = 0.0f;
}

__global__ void k_copy2d(float* __restrict__ dst, int dld,
                         const float* __restrict__ src, int sld,
                         int rows, int cols)
{
    int i = blockIdx.x * blockDim.x + threadIdx.x;
    if (i >= rows * cols) return;
    int r = i / cols, c = i - r * cols;
    dst[(size_t)r * dld + c] = src[(size_t)r * sld + c];
}

__global__ void k_embed(const int* __restrict__ ids,
                        const float* __restrict__ emb,
                        float* __restrict__ out, int rows, int dim)
{
    int i = blockIdx.x * blockDim.x + threadIdx.x;
    if (i >= rows * dim) return;
    int r = i / dim, c = i - r * dim;
    out[i] = emb[(size_t)ids[r] * dim + c];
}

// xc = [char_emb[ids] (256) | pos_emb[pos] (64) | char_ctx (1024)]  -> (B,1344)
__global__ void k_build_xc(const int* __restrict__ ids,
                           const float* __restrict__ emb,
                           const float* __restrict__ pe,
                           const float* __restrict__ cctx,
                           float* __restrict__ xc)
{
    int i = blockIdx.x * blockDim.x + threadIdx.x;
    if (i >= BB * 1344) return;
    int b = i / 1344, c = i - b * 1344;
    float v;
    if (c < EMB)             v = emb[(size_t)ids[b] * EMB + c];
    else if (c < EMB + POSD) v = pe[c - EMB];
    else                     v = cctx[(size_t)b * H2 + (c - EMB - POSD)];
    xc[i] = v;
}

// Luong dot attention for one batch element per block (256 = 32 rows x 8 threads).
// enc_out: (L,B,1024) ; ctx: (B,1024) ; att -> att[b*att_ld + j]
__global__ __launch_bounds__(256)
void k_attention(const float* __restrict__ enc_out,
                 const float* __restrict__ ctx,
                 float* __restrict__ att, int att_ld)
{
    const int b = blockIdx.x;
    __shared__ float part[LSEQ][8];
    __shared__ float sw[LSEQ];
    const int tid = threadIdx.x;
    const int t8 = tid >> 3, l8 = tid & 7;

    {   // partial dot products: 8 threads per sequence position
        const float* e  = enc_out + ((size_t)t8 * BB + b) * H2;
        const float* cx = ctx + (size_t)b * H2;
        float s = 0.0f;
        for (int j = l8; j < H2; j += 8) s += e[j] * cx[j];
        part[t8][l8] = s;
    }
    __syncthreads();
    if (tid < LSEQ) {
        float s = 0.0f;
#pragma unroll
        for (int k = 0; k < 8; ++k) s += part[tid][k];
        sw[tid] = s;
    }
    __syncthreads();
    if (tid == 0) {
        float mx = -INFINITY;
        for (int t = 0; t < LSEQ; ++t) mx = fmaxf(mx, sw[t]);
        float den = 0.0f;
        for (int t = 0; t < LSEQ; ++t) { float e = expf(sw[t] - mx); sw[t] = e; den += e; }
        const float inv = 1.0f / den;
        for (int t = 0; t < LSEQ; ++t) sw[t] *= inv;
    }
    __syncthreads();
    for (int j = tid; j < H2; j += blockDim.x) {
        float s = 0.0f;
        for (int t = 0; t < LSEQ; ++t)
            s += sw[t] * enc_out[((size_t)t * BB + b) * H2 + j];
        att[(size_t)b * att_ld + j] = s;
    }
}

// in-place log_softmax over the last dim; one block per row.
__global__ __launch_bounds__(256)
void k_log_softmax(float* __restrict__ x, int cols)
{
    float* p = x + (size_t)blockIdx.x * cols;
    __shared__ float red[256];
    const int tid = threadIdx.x;

    float mx = -INFINITY;
    for (int j = tid; j < cols; j += 256) mx = fmaxf(mx, p[j]);
    red[tid] = mx; __syncthreads();
    for (int s = 128; s > 0; s >>= 1) {
        if (tid < s) red[tid] = fmaxf(red[tid], red[tid + s]);
        __syncthreads();
    }
    mx = red[0]; __syncthreads();

    float sum = 0.0f;
    for (int j = tid; j < cols; j += 256) sum += expf(p[j] - mx);
    red[tid] = sum; __syncthreads();
    for (int s = 128; s > 0; s >>= 1) {
        if (tid < s) red[tid] += red[tid + s];
        __syncthreads();
    }
    const float lse = mx + logf(red[0]);
    __syncthreads();
    for (int j = tid; j < cols; j += 256) p[j] -= lse;
}

// ---------------- host-side helpers ----------------
static inline void gemm2(hipStream_t s,
                         const float* A, int lda, int K,
                         const float* W, int ldw,
                         const float* A2, int lda2, int K2,
                         const float* W2, int ldw2,
                         const float* bias, float* C, int ldc,
                         int M, int N, int acc)
{
    dim3 grid((N + 255) / 256, (M + 15) / 16);
    k_gemm<<<grid, 128, 0, s>>>(A, lda, K, W, ldw, A2, lda2, K2, W2, ldw2,
                                bias, C, ldc, M, N, acc);
}
static inline void gemm(hipStream_t s, const float* A, int lda, int K,
                        const float* W, int ldw, const float* bias,
                        float* C, int ldc, int M, int N, int acc)
{
    gemm2(s, A, lda, K, W, ldw, nullptr, 0, 0, nullptr, 0, bias, C, ldc, M, N, acc);
}
static inline void fill0(hipStream_t s, float* p, int n)
{
    k_fill0<<<(n + 255) / 256, 256, 0, s>>>(p, n);
}
static inline void copy2d(hipStream_t s, float* dst, int dld,
                          const float* src, int sld, int rows, int cols)
{
    k_copy2d<<<(rows * cols + 255) / 256, 256, 0, s>>>(dst, dld, src, sld, rows, cols);
}

extern "C" void kernel_launch(void* const* d_in, const int* in_sizes, int n_in,
                              void* d_out, int out_size, void* d_ws, size_t ws_size,
                              hipStream_t stream)
{
    (void)in_sizes; (void)n_in; (void)out_size; (void)ws_size;

    // ---- inputs: jax pytree (sorted-key) flatten of setup_inputs() ----
    const float* Wout  = (const float*)d_in[0];   // (V, 2048)
    const float* bout  = (const float*)d_in[1];   // (V,)
    const float* cemb  = (const float*)d_in[2];   // (V, 256)
    const float* chWi0 = (const float*)d_in[3];   // (4096, 1344)
    const float* chWh0 = (const float*)d_in[4];   // (4096, 1024)
    const float* chb0  = (const float*)d_in[5];
    const float* chWi1 = (const float*)d_in[6];   // (4096, 1024)
    const float* chWh1 = (const float*)d_in[7];
    const float* chb1  = (const float*)d_in[8];
    const float* eWi[NLAY][2], *eWh[NLAY][2], *eb[NLAY][2];
    for (int l = 0; l < NLAY; ++l)
        for (int d = 0; d < 2; ++d) {
            int base = 9 + (l * 2 + d) * 3;
            eWi[l][d] = (const float*)d_in[base + 0]; // (2048, in)
            eWh[l][d] = (const float*)d_in[base + 1]; // (2048, 512)
            eb [l][d] = (const float*)d_in[base + 2]; // (2048,)
        }
    const float* pemb  = (const float*)d_in[21];  // (28, 64)
    const float* sWi0  = (const float*)d_in[22];  // (4096, 2048)
    const float* sWh0  = (const float*)d_in[23];  // (4096, 1024)
    const float* sb0   = (const float*)d_in[24];
    const float* sWi1  = (const float*)d_in[25];  // (4096, 1024)
    const float* sWh1  = (const float*)d_in[26];
    const float* sb1   = (const float*)d_in[27];
    const int* sentences = (const int*)d_in[28];  // (32, 128)
    const int* targets   = (const int*)d_in[29];  // (28, 128)
    float* out = (float*)d_out;                   // (28, 128, 12000)

    // ---- workspace arena (floats) ----
    float* ws = (float*)d_ws;
    float* bufA  = ws;                               // (L*B, 1024) emb / layer1 out
    float* bufB  = bufA  + (size_t)LSEQ * BB * H2;   // (L*B, 1024) layer0 out
    float* xW    = bufB  + (size_t)LSEQ * BB * H2;   // (L*B, 2048) gate pre-acts
    float* hbuf  = xW    + (size_t)LSEQ * BB * H4;   // (B, 512)
    float* cbuf  = hbuf  + BB * HH;                  // (B, 512)
    float* dec_h = cbuf  + BB * HH;                  // (2, B, 1024)
    float* dec_c = dec_h + 2 * BB * H2;              // (2, B, 1024)
    float* chh   = dec_c + 2 * BB * H2;              // (2, B, 1024)
    float* chc   = chh   + 2 * BB * H2;              // (2, B, 1024)
    float* zbuf  = chc   + 2 * BB * H2;              // (B, 4096)
    float* xsbuf = zbuf  + BB * 4 * H2;              // (B, 2048) xs / xc / [ctx|att]
    float* prevs = xsbuf + BB * H4;                  // (B, 1024) prev_sent
    float* cctx  = prevs + BB * H2;                  // (B, 1024) context

    // ================= encoder =================
    k_embed<<<(LSEQ * BB * EMB + 255) / 256, 256, 0, stream>>>(
        sentences, cemb, bufA, LSEQ * BB, EMB);

    for (int l = 0; l < NLAY; ++l) {
        const float* xin = (l == 0) ? bufA : bufB;
        float* xout      = (l == 0) ? bufB : bufA;
        const int kin    = (l == 0) ? EMB : H2;
        for (int d = 0; d < 2; ++d) {
            // hoisted input projection for all timesteps: xW = x @ Wih^T + b
            gemm(stream, xin, kin, kin, eWi[l][d], kin, eb[l][d],
                 xW, H4, LSEQ * BB, H4, 0);
            fill0(stream, hbuf, BB * HH);
            fill0(stream, cbuf, BB * HH);
            for (int step = 0; step < LSEQ; ++step) {
                const int t = d ? (LSEQ - 1 - step) : step;
                float* zt = xW + (size_t)t * BB * H4;
                gemm(stream, hbuf, HH, HH, eWh[l][d], HH, nullptr,
                     zt, H4, BB, H4, 1);
                k_cell<<<(BB * HH + 255) / 256, 256, 0, stream>>>(
                    zt, H4, cbuf, hbuf, cbuf, HH,
                    xout + (size_t)t * BB * H2 + (d ? HH : 0), H2);
            }
            // capture final h,c into decoder state halves
            copy2d(stream, dec_h + (size_t)l * BB * H2 + (d ? HH : 0), H2,
                   hbuf, HH, BB, HH);
            copy2d(stream, dec_c + (size_t)l * BB * H2 + (d ? HH : 0), H2,
                   cbuf, HH, BB, HH);
        }
    }
    const float* enc_out = bufA;                                    // (L,B,1024)
    copy2d(stream, cctx, H2, dec_h + (size_t)BB * H2, H2, BB, H2);  // context = h[-1]
    copy2d(stream, prevs, H2, cctx, H2, BB, H2);                    // prev_sent = context

    // ================= decoder =================
    for (int ts = 0; ts < NSENT; ++ts) {
        // ---- sentence cell: xs = [context | prev_sent] ----
        copy2d(stream, xsbuf, H4, cctx, H2, BB, H2);
        copy2d(stream, xsbuf + H2, H4, prevs, H2, BB, H2);
        // layer 0 (fused: z = xs@Wi0^T + h0@Wh0^T + b0)
        gemm2(stream, xsbuf, H4, H4, sWi0, H4, dec_h, H2, H2, sWh0, H2,
              sb0, zbuf, 4 * H2, BB, 4 * H2, 0);
        k_cell<<<(BB * H2 + 255) / 256, 256, 0, stream>>>(
            zbuf, 4 * H2, dec_c, dec_h, dec_c, H2, nullptr, 0);
        // layer 1
        gemm2(stream, dec_h, H2, H2, sWi1, H2,
              dec_h + (size_t)BB * H2, H2, H2, sWh1, H2,
              sb1, zbuf, 4 * H2, BB, 4 * H2, 0);
        k_cell<<<(BB * H2 + 255) / 256, 256, 0, stream>>>(
            zbuf, 4 * H2, dec_c + (size_t)BB * H2,
            dec_h + (size_t)BB * H2, dec_c + (size_t)BB * H2, H2, nullptr, 0);
        // char decoder state = copy of sentence state
        copy2d(stream, chh, H2, dec_h, H2, 2 * BB, H2);
        copy2d(stream, chc, H2, dec_c, H2, 2 * BB, H2);
        const float* char_ctx = dec_h + (size_t)BB * H2;   // fixed during char loop

        for (int tc = 0; tc < NCHR; ++tc) {
            const int pos = ts * NCHR + tc;
            const int* ids = (pos == 0) ? (sentences + (size_t)(LSEQ - 1) * BB)
                                        : (targets + (size_t)(pos - 1) * BB);
            // xc = [ce | pe | char_ctx] -> (B,1344)
            k_build_xc<<<(BB * 1344 + 255) / 256, 256, 0, stream>>>(
                ids, cemb, pemb + (size_t)pos * POSD, char_ctx, xsbuf);
            // char layer 0 (fused)
            gemm2(stream, xsbuf, 1344, 1344, chWi0, 1344, chh, H2, H2, chWh0, H2,
                  chb0, zbuf, 4 * H2, BB, 4 * H2, 0);
            k_cell<<<(BB * H2 + 255) / 256, 256, 0, stream>>>(
                zbuf, 4 * H2, chc, chh, chc, H2, nullptr, 0);
            // char layer 1 (fused)
            gemm2(stream, chh, H2, H2, chWi1, H2,
                  chh + (size_t)BB * H2, H2, H2, chWh1, H2,
                  chb1, zbuf, 4 * H2, BB, 4 * H2, 0);
            k_cell<<<(BB * H2 + 255) / 256, 256, 0, stream>>>(
                zbuf, 4 * H2, chc + (size_t)BB * H2,
                chh + (size_t)BB * H2, chc + (size_t)BB * H2, H2, nullptr, 0);
            const float* ctx = chh + (size_t)BB * H2;
            // prev_sent = ctx
            copy2d(stream, prevs, H2, ctx, H2, BB, H2);
            // proj input = [ctx | att] into xsbuf (B,2048)
            copy2d(stream, xsbuf, H4, ctx, H2, BB, H2);
            k_attention<<<BB, 256, 0, stream>>>(enc_out, ctx, xsbuf + H2, H4);
            // logits for this step -> d_out slice
            gemm(stream, xsbuf, H4, H4, Wout, H4, bout,
                 out + (size_t)pos * BB * VV, VV, BB, VV, 0);
        }
    }

    // final log_softmax over vocab, in place on d_out
    k_log_softmax<<<NSENT * NCHR * BB, 256, 0, stream>>>(out, VV);
}